// PaddingJacobiens_38766374813842
// MI455X (gfx1250) — compile-verified
//
#include <hip/hip_runtime.h>

// PaddingJacobiens: B=32, H=W=64, C=64, K=3, S=1, SAME pad -> Hp=Wp=66,
// out spatial 192x192. The reference's selector-matrix einsums + raw reshapes
// collapse to a pure gather:
//   out_flat[b, i*36864 + r*192 + s] = upflat[b, i*4356 + (r/3 + r%3)*66 + (s/3 + s%3)]
//   upflat[b, L] = in[b, rr-1, qq-1, c]  (p=L>>6, c=L&63, rr=p/66, qq=p%66), 0 on border.
// Windows [i*4356, (i+1)*4356) partition upflat exactly once per batch ->
// stage each window in LDS via async-to-LDS B128 DMA; HBM reads each input
// byte exactly once; output streams out as 16B non-temporal stores.

#define TPB 256
#define WIN 4356        // 66*66 window elements per (b,i)
#define CHUNKS 1089     // WIN/4: 16-byte staging chunks
#define GROUPS 3072     // 192 rows * 16 twelve-wide output groups

typedef float vf4 __attribute__((ext_vector_type(4)));   // clang vector: valid for NT builtins

__global__ __launch_bounds__(TPB) void padjac_gather_kernel(
    const float* __restrict__ in, float* __restrict__ out)
{
    __shared__ __align__(16) float win[WIN];

    const int tid = threadIdx.x;
    const int blk = blockIdx.x;      // b*64 + i
    const int b   = blk >> 6;
    const int i   = blk & 63;

    const unsigned long long inBase  = (unsigned long long)b * 262144ull;  // 64*64*64
    const unsigned long long outBase = (unsigned long long)blk * 36864ull; // 192*192

    // 32-bit LDS byte address of win[0] (flat->LDS truncation per ISA aperture rules)
    const unsigned winAddr = (unsigned)(unsigned long long)(&win[0]);

    // ---- Stage the upflat window into LDS, 16B at a time ----
    // A 4-float chunk (L % 4 == 0) never straddles an up-row (64 % 4 == 0),
    // so each chunk is uniformly interior (async B128 DMA, tracked by
    // ASYNCcnt) or border (16B zero fill).
    for (int ch = tid; ch < CHUNKS; ch += TPB) {
        const int e  = ch << 2;
        const int L  = i * WIN + e;
        const int p  = L >> 6;
        const int c  = L & 63;              // multiple of 4
        const int rr = p / 66;
        const int qq = p - rr * 66;
        if (rr >= 1 && rr <= 64 && qq >= 1 && qq <= 64) {
            const float* src =
                in + inBase + ((unsigned long long)(((rr - 1) << 6) + (qq - 1)) << 6) + c;
            const unsigned           ldsDst = winAddr + ((unsigned)ch << 4);
            const unsigned long long gAddr  = (unsigned long long)src;
            asm volatile("global_load_async_to_lds_b128 %0, %1, off"
                         :: "v"(ldsDst), "v"(gAddr)
                         : "memory");
        } else {
            vf4 z = {0.f, 0.f, 0.f, 0.f};
            *(vf4*)&win[e] = z;             // 16B-aligned ds_store_b128
        }
    }
    // Wave-level completion of async DMA, then workgroup barrier so every
    // wave observes the fully staged window.
    asm volatile("s_wait_asynccnt 0x0" ::: "memory");
    __syncthreads();

    // ---- Emit output ----
    // For s = 12t..12t+11: k = 4t + [0,1,2, 1,2,3, 2,3,4, 3,4,5] -> twelve
    // outputs are permutations of six contiguous LDS floats. Non-temporal
    // B128 stores: output is write-once streaming (302 MB); keep L2 for the
    // resident data.
    for (int g = tid; g < GROUPS; g += TPB) {
        const int r = g >> 4;               // output row 0..191
        const int t = g & 15;               // 12-wide group within row
        const int j = r / 3 + r % 3;        // padded-frame row select
        const int base = j * 66 + (t << 2);

        const float w0 = win[base + 0];
        const float w1 = win[base + 1];
        const float w2 = win[base + 2];
        const float w3 = win[base + 3];
        const float w4 = win[base + 4];
        const float w5 = win[base + 5];

        float* dst = out + outBase + (unsigned long long)r * 192ull + 12u * t;
        const vf4 q0 = {w0, w1, w2, w1};    // d = 0..3
        const vf4 q1 = {w2, w3, w2, w3};    // d = 4..7
        const vf4 q2 = {w4, w3, w4, w5};    // d = 8..11
        __builtin_nontemporal_store(q0, (vf4*)(dst + 0));
        __builtin_nontemporal_store(q1, (vf4*)(dst + 4));
        __builtin_nontemporal_store(q2, (vf4*)(dst + 8));
    }
}

extern "C" void kernel_launch(void* const* d_in, const int* in_sizes, int n_in,
                              void* d_out, int out_size, void* d_ws, size_t ws_size,
                              hipStream_t stream)
{
    (void)in_sizes; (void)n_in; (void)d_ws; (void)ws_size; (void)out_size;
    const float* in  = (const float*)d_in[0];
    float*       out = (float*)d_out;

    // One block per (batch, i) window: 32 * 64 = 2048 blocks, 256 threads each.
    padjac_gather_kernel<<<dim3(32 * 64), dim3(TPB), 0, stream>>>(in, out);
}